// LinearPoolNewsvendorLayer_36902359007816
// MI455X (gfx1250) — compile-verified
//
#include <hip/hip_runtime.h>

typedef __attribute__((ext_vector_type(2))) float        v2f;
typedef __attribute__((ext_vector_type(8))) float        v8f;
typedef __attribute__((ext_vector_type(4))) unsigned int v4u;
typedef __attribute__((ext_vector_type(8))) int          v8i;
typedef __attribute__((ext_vector_type(4))) int          v4i;

#define M_INPUTS 8
#define B_ROWS   8192
#define L_LOC    1024
#define TAU_C    0.5f
#define RA_C     0.2f
#define EPS_C    1e-4f
#define N_BISECT 64

// LDS plane stride in floats: 1024 + 16 pad (TDM pad: 4 DWORDs per 256 DWORDs).
// Plane k vs k+2 differ by 2080 words == 32 mod 64 banks -> conflict-free
// dual-half WMMA B gathers.
#define XPITCH   1040

// One workgroup per batch row b. 256 threads = 8 wave32.
// Phase 0: TDM tensor_load_to_lds DMAs the [8 x 1024] f32 row-block into LDS.
// Phase 1: combined_pdf row via f32 WMMA (K=8 as two 16x16x4 f32 WMMAs).
// Phase 2: CDF prefix-scan + q-moment reductions (once per row).
// Phase 3: 64-iteration bisection with O(log L) gradient evals (CDF lookup).
// Phase 4: z and error = support - z written out.
__global__ __launch_bounds__(256) void newsvendor_kernel(
    const float* __restrict__ X,      // [M, B, L]
    const float* __restrict__ W,      // [M]
    const float* __restrict__ S,      // [L]
    float* __restrict__ out_comb,     // [B, L]
    float* __restrict__ out_z,        // [B]
    float* __restrict__ out_err)      // [B, L]
{
  __shared__ float s_x[M_INPUTS * XPITCH];   // TDM-staged input tile (padded)
  __shared__ float s_sup[L_LOC];
  __shared__ float s_p[L_LOC];
  __shared__ float s_cdf[L_LOC];
  __shared__ float s_scan[256];
  __shared__ float s_r0[256];
  __shared__ float s_r1[256];

  const int t    = threadIdx.x;
  const int b    = blockIdx.x;
  const int lane = t & 31;
  const int wave = t >> 5;
  const size_t BL = (size_t)B_ROWS * L_LOC;

  // support row -> LDS (coalesced float4)
  {
    const int l = t * 4;
    *(float4*)&s_sup[l] = *(const float4*)&S[l];
  }

  // ---- Phase 0: TDM DMA of X[:, b, :] (8 x 1024 f32, stride B*L) into LDS ----
  if (wave == 0) {
    const unsigned long long ga =
        (unsigned long long)(uintptr_t)(X + (size_t)b * L_LOC);
    const unsigned int lds_base = (unsigned int)(uintptr_t)&s_x[0];

    v4u g0;
    g0[0] = 1u;                                   // count=1, is_restore=0
    g0[1] = lds_base;                             // lds_addr (bytes)
    g0[2] = (unsigned int)(ga & 0xFFFFFFFFull);   // global_addr[31:0]
    g0[3] = (unsigned int)((ga >> 32) & 0x1FFFFFFull) | (2u << 30); // addr[56:32], type=2

    v8i g1;
    g1[0] = (int)((2u << 16)      // data_size = 4B
                | (1u << 20)      // pad_enable
                | (7u << 22)      // pad_interval = 256 DWORDs
                | (3u << 25));    // pad_amount  = 4 DWORDs
    g1[1] = (int)((unsigned)L_LOC << 16);         // tensor_dim0[15:0] in [63:48]
    g1[2] = (int)((unsigned)M_INPUTS << 16);      // tensor_dim1[15:0] in [95:80]
    g1[3] = (int)((unsigned)L_LOC << 16);         // tile_dim0 in [127:112]
    g1[4] = (int)M_INPUTS;                        // tile_dim1 in [143:128]
    g1[5] = (int)BL;                              // tensor_dim0_stride[31:0]
    g1[6] = 0;                                    // stride[47:32], dim1_stride lo
    g1[7] = 0;

    v4i g2 = {0, 0, 0, 0};                        // 2-D tile: groups 2/3 unused
    v4i g3 = {0, 0, 0, 0};
    v8i g4 = {0, 0, 0, 0, 0, 0, 0, 0};            // extra group (clang-23 form)

    __builtin_amdgcn_tensor_load_to_lds(g0, g1, g2, g3, g4, 0);
    __builtin_amdgcn_s_wait_tensorcnt((short)0);
  }

  // weights (uniform loads)
  const float w0 = W[0], w1 = W[1], w2 = W[2], w3 = W[3];
  const float w4 = W[4], w5 = W[5], w6 = W[6], w7 = W[7];

  // A-matrix (16x4 f32): lanes 0-15 = rows (all identical = weights),
  // VGPR0 holds K=0 (lanes 0-15) / K=2 (lanes 16-31), VGPR1 holds K=1 / K=3.
  const int mo = lane >> 4;
  const int n  = lane & 15;
  v2f aLo; aLo.x = mo ? w2 : w0; aLo.y = mo ? w3 : w1;
  v2f aHi; aHi.x = mo ? w6 : w4; aHi.y = mo ? w7 : w5;

  __syncthreads();   // TDM tile visible to all waves

  // ---- Phase 1: combine via WMMA; B operands gathered from padded LDS ----
  const int k0 = mo ? 2 : 0;   // K plane for B VGPR0 lane-half
  for (int g = 0; g < 8; ++g) {
    const int j0 = wave * 128 + g * 16;
    const int cp = j0 + ((j0 >> 8) << 2) + n;    // padded column index
    v2f bLo, bHi;
    bLo.x = s_x[(k0 + 0) * XPITCH + cp];
    bLo.y = s_x[(k0 + 1) * XPITCH + cp];
    bHi.x = s_x[(k0 + 4) * XPITCH + cp];
    bHi.y = s_x[(k0 + 5) * XPITCH + cp];
    v8f c = {};
    c = __builtin_amdgcn_wmma_f32_16x16x4_f32(false, aLo, false, bLo,
                                              (short)0, c, false, false);
    c = __builtin_amdgcn_wmma_f32_16x16x4_f32(false, aHi, false, bHi,
                                              (short)0, c, false, false);
    if (lane < 16) s_p[j0 + lane] = c[0];
  }
  __syncthreads();

  // ---- Phase 2: coalesced combined_pdf write; q-moments; CDF scan ----
  const int l0 = t * 4;
  const float4 p4 = *(const float4*)&s_p[l0];
  *(float4*)&out_comb[(size_t)b * L_LOC + l0] = p4;

  float q0 = sqrtf(p4.x + EPS_C); q0 *= q0;
  float q1 = sqrtf(p4.y + EPS_C); q1 *= q1;
  float q2 = sqrtf(p4.z + EPS_C); q2 *= q2;
  float q3 = sqrtf(p4.w + EPS_C); q3 *= q3;
  const float su0 = s_sup[l0+0], su1 = s_sup[l0+1];
  const float su2 = s_sup[l0+2], su3 = s_sup[l0+3];
  s_r0[t] = q0 + q1 + q2 + q3;                   // Sum q
  s_r1[t] = q0*su0 + q1*su1 + q2*su2 + q3*su3;   // Sum q*s

  const float c0 = p4.x, c1 = c0 + p4.y, c2 = c1 + p4.z, c3 = c2 + p4.w;
  s_scan[t] = c3;
  __syncthreads();

  for (int off = 128; off > 0; off >>= 1) {
    if (t < off) { s_r0[t] += s_r0[t + off]; s_r1[t] += s_r1[t + off]; }
    __syncthreads();
  }
  const float Sq0 = s_r0[0];
  const float Sq1 = s_r1[0];

  for (int off = 1; off < 256; off <<= 1) {
    const float v   = s_scan[t];
    const float add = (t >= off) ? s_scan[t - off] : 0.f;
    __syncthreads();
    s_scan[t] = v + add;
    __syncthreads();
  }
  const float pre = (t > 0) ? s_scan[t - 1] : 0.f;
  s_cdf[l0+0] = pre + c0;
  s_cdf[l0+1] = pre + c1;
  s_cdf[l0+2] = pre + c2;
  s_cdf[l0+3] = pre + c3;
  __syncthreads();

  const float S_all = s_cdf[L_LOC - 1];
  const float z_max = s_sup[L_LOC - 1];

  // ---- Phase 3: bisection (uniform control flow, broadcast LDS reads) ----
  float blo = 0.f, bhi = z_max;
  for (int it = 0; it < N_BISECT; ++it) {
    const float mid = 0.5f * (blo + bhi);
    int k = 0;
#pragma unroll
    for (int stp = 512; stp > 0; stp >>= 1) {
      const int cand = k + stp;
      if (cand <= L_LOC && s_sup[cand - 1] <= mid) k = cand;
    }
    const float F    = (k > 0) ? s_cdf[k - 1] : 0.f;
    const float pin  = (1.0f - TAU_C) * F - TAU_C * (S_all - F);
    const float quad = -4.0f * RA_C * (Sq1 - mid * Sq0);
    const float gval = 2.0f * (1.0f - RA_C) * pin + quad;
    const bool right = gval < 0.f;
    blo = right ? mid : blo;
    bhi = right ? bhi : mid;
  }
  const float z = 0.5f * (blo + bhi);
  if (t == 0) out_z[b] = z;

  // ---- Phase 4: error = support - z ----
  float4 e4;
  e4.x = su0 - z; e4.y = su1 - z; e4.z = su2 - z; e4.w = su3 - z;
  *(float4*)&out_err[(size_t)b * L_LOC + l0] = e4;
}

extern "C" void kernel_launch(void* const* d_in, const int* in_sizes, int n_in,
                              void* d_out, int out_size, void* d_ws, size_t ws_size,
                              hipStream_t stream) {
  const float* X = (const float*)d_in[0];   // list_inputs [M,B,L]
  const float* W = (const float*)d_in[1];   // weights [M]
  const float* S = (const float*)d_in[2];   // support [L]
  float* out      = (float*)d_out;
  float* out_comb = out;                                  // [B,L]
  float* out_z    = out + (size_t)B_ROWS * L_LOC;         // [B]
  float* out_err  = out_z + B_ROWS;                       // [B,L]
  hipLaunchKernelGGL(newsvendor_kernel, dim3(B_ROWS), dim3(256), 0, stream,
                     X, W, S, out_comb, out_z, out_err);
}